// AdjacentAttention_51101520888593
// MI455X (gfx1250) — compile-verified
//
#include <hip/hip_runtime.h>
#include <math.h>
#include <float.h>

// MI455X / gfx1250, wave32. Matrix ops: V_WMMA_F32_16X16X4_F32 (fp32 in/out).
typedef __attribute__((ext_vector_type(2))) float v2f;
typedef __attribute__((ext_vector_type(8))) float v8f;
typedef __attribute__((ext_vector_type(4))) int   v4i;
typedef __attribute__((ext_vector_type(2))) int   v2i;

#define DD   256   // model dim D
#define HH   4     // heads
#define DHH  64    // head dim
#define AA   32    // neighbors
#define KVW  512   // 2*H*DH (kv row width)

#define KSLAB   16   // K rows per LDS slab
#define WSTRIDE 272  // padded LDS row stride (floats) for W slab: 2*272 % 64 == 32
#define ASTRIDE 18   // padded LDS row stride (floats) for A slab

// address-space + pointee casts for the async-to-LDS builtins
// (builtin wants: arg0 = vNi __device__* (AS1, non-const), arg1 = vNi __shared__* (AS3))
#define G128(p) ((__attribute__((address_space(1))) v4i*)(p))
#define G64(p)  ((__attribute__((address_space(1))) v2i*)(p))
#define L128(p) ((__attribute__((address_space(3))) v4i*)(p))
#define L64(p)  ((__attribute__((address_space(3))) v2i*)(p))

// ---------------------------------------------------------------------------
// fp32 WMMA GEMM: C[N x M] = A[N x K] @ W[K x M] (+ bias), row major.
// Block = 128 threads (4 waves) -> 16 rows x 256 cols of C; each wave owns a
// 16x64 stripe (4 f32 accumulators, A fragment reused across the 4 tiles).
// K is consumed in 16-row slabs staged in LDS by GLOBAL_LOAD_ASYNC_TO_LDS_*,
// double buffered: slab s+1 streams in (ASYNCcnt) while slab s feeds WMMA.
// Every thread issues exactly 9 async ops/slab (8 x b128 W, 1 x b64 A), so
// s_wait_asynccnt 9 == "previous slab fully landed, prefetch still in flight".
// ---------------------------------------------------------------------------
__global__ __launch_bounds__(128) void gemm_wmma_f32(
    const float* __restrict__ A, const float* __restrict__ W,
    const float* __restrict__ bias, float* __restrict__ C,
    int Nrows, int K, int M)
{
  __shared__ __align__(16) float Wl[2][KSLAB * WSTRIDE];
  __shared__ __align__(16) float Al[2][16 * ASTRIDE];

  const int tid     = threadIdx.x;
  const int lane    = tid & 31;
  const int wave    = tid >> 5;
  const int colbase = blockIdx.y * 256;          // block covers 256 cols
  const int row0    = blockIdx.x * 16;
  const int kh      = lane >> 4;                 // K-pair select {0,1} vs {2,3}
  const int ml      = lane & 15;

  // per-thread async-copy assignment for the A slab (fixed across slabs)
  int arow = row0 + (tid >> 3);                  // thread -> (row, 2-float chunk)
  if (arow >= Nrows) arow = Nrows - 1;
  const int aj = (tid & 7) * 2;

  auto issue_slab = [&](int k0, int buf) {
#pragma unroll
    for (int i = 0; i < 8; ++i) {                // W slab: 1024 b128 chunks / 128 thr
      const int chunk = i * 128 + tid;
      const int r = chunk >> 6;
      const int c = (chunk & 63) * 4;
      __builtin_amdgcn_global_load_async_to_lds_b128(
          G128(W + (size_t)(k0 + r) * M + colbase + c),
          L128(&Wl[buf][r * WSTRIDE + c]), 0, 0);
    }
    __builtin_amdgcn_global_load_async_to_lds_b64(
        G64(A + (size_t)arow * K + k0 + aj),
        L64(&Al[buf][(tid >> 3) * ASTRIDE + aj]), 0, 0);
  };

  v8f acc0 = {}, acc1 = {}, acc2 = {}, acc3 = {};
  const int NS = K / KSLAB;
  const int wc = wave * 64 + ml;                 // wave's column within LDS slab

  issue_slab(0, 0);
  for (int s = 0; s < NS; ++s) {
    const int buf = s & 1;
    if (s + 1 < NS) {
      issue_slab((s + 1) * KSLAB, buf ^ 1);      // prefetch next slab
      __builtin_amdgcn_s_wait_asynccnt(9);       // slab s landed; s+1 in flight
    } else {
      __builtin_amdgcn_s_wait_asynccnt(0);
    }
    __syncthreads();

    const float* __restrict__ wl = &Wl[buf][0];
    const float* __restrict__ al = &Al[buf][0];
#pragma unroll
    for (int kk = 0; kk < KSLAB; kk += 4) {
      const int r0 = kk + 2 * kh;
      v2f a = *(const v2f*)&al[ml * ASTRIDE + r0];
      const float* __restrict__ wr0 = &wl[r0 * WSTRIDE + wc];
      const float* __restrict__ wr1 = wr0 + WSTRIDE;
      // pull all B words first so ds latency overlaps, then 4 back-to-back WMMAs
      const float bx0 = wr0[0],  by0 = wr1[0];
      const float bx1 = wr0[16], by1 = wr1[16];
      const float bx2 = wr0[32], by2 = wr1[32];
      const float bx3 = wr0[48], by3 = wr1[48];
      v2f b;
      b.x = bx0; b.y = by0;
      acc0 = __builtin_amdgcn_wmma_f32_16x16x4_f32(false, a, false, b, (short)0, acc0, false, false);
      b.x = bx1; b.y = by1;
      acc1 = __builtin_amdgcn_wmma_f32_16x16x4_f32(false, a, false, b, (short)0, acc1, false, false);
      b.x = bx2; b.y = by2;
      acc2 = __builtin_amdgcn_wmma_f32_16x16x4_f32(false, a, false, b, (short)0, acc2, false, false);
      b.x = bx3; b.y = by3;
      acc3 = __builtin_amdgcn_wmma_f32_16x16x4_f32(false, a, false, b, (short)0, acc3, false, false);
    }
    __syncthreads();                             // all waves done with buf before reuse
  }

  // D layout: reg r -> row = row0 + r + 8*kh, col = colbase + wave*64 + t*16 + ml
  const int colb = colbase + wave * 64 + ml;
  const int rowb = row0 + 8 * kh;
  const float b0 = bias ? bias[colb +  0] : 0.f;
  const float b1 = bias ? bias[colb + 16] : 0.f;
  const float b2 = bias ? bias[colb + 32] : 0.f;
  const float b3 = bias ? bias[colb + 48] : 0.f;

#pragma unroll
  for (int r = 0; r < 8; ++r) {
    const int row = rowb + r;
    if (row < Nrows) {
      float* __restrict__ crow = C + (size_t)row * M + colb;
      crow[ 0] = acc0[r] + b0;
      crow[16] = acc1[r] + b1;
      crow[32] = acc2[r] + b2;
      crow[48] = acc3[r] + b3;
    }
  }
}

// ---------------------------------------------------------------------------
// Adjacent attention over precomputed Q (N x 256) and KV (N x 512, k|v).
// One wave per (token n, head h): lane a owns neighbor a; null slot uniform.
// Phase 1: per-lane sim = q.k_a ; wave softmax over {null, 32 lanes}.
// Phase 2: probs+indices through LDS; lane owns dims {2*lane, 2*lane+1} and
//          accumulates attn . v (each v row read once).
// out may alias Q: the wave fully consumes its Q slice before writing it.
// ---------------------------------------------------------------------------
__global__ __launch_bounds__(128) void adj_attention(
    const float* __restrict__ Q, const float* __restrict__ KV,
    const int* __restrict__ adj, const unsigned char* __restrict__ mask,
    const float* __restrict__ null_k, const float* __restrict__ null_v,
    float* __restrict__ out, int N)
{
  __shared__ float sp[HH][AA];
  __shared__ int   si[HH][AA];

  const int n    = blockIdx.x;
  const int h    = threadIdx.x >> 5;     // wave id = head
  const int lane = threadIdx.x & 31;
  if (n >= N) return;

  const int  idx = adj[(size_t)n * AA + lane];
  const bool mk  = mask[(size_t)n * AA + lane] != 0;

  const float* __restrict__ qrow = Q  + (size_t)n   * DD  + h * DHH;  // wave-uniform
  const float* __restrict__ krow = KV + (size_t)idx * KVW + h * DHH;  // per-lane gather
  const float* __restrict__ nk   = null_k + h * DHH;
  const float* __restrict__ nv   = null_v + h * DHH;

  float sim = 0.f, simn = 0.f;
#pragma unroll 8
  for (int d = 0; d < DHH; ++d) {
    const float qd = qrow[d];
    sim  += qd * krow[d];
    simn += qd * nk[d];
  }
  const float scale = 0.125f;            // DH^-0.5 = 64^-0.5
  sim  *= scale;
  simn *= scale;
  if (!mk) sim = -FLT_MAX;

  // wave-wide max over 32 lanes, then fold in the (uniform) null slot
  float mx = sim;
#pragma unroll
  for (int off = 16; off >= 1; off >>= 1)
    mx = fmaxf(mx, __shfl_xor(mx, off, 32));
  mx = fmaxf(mx, simn);

  float p  = mk ? __expf(sim - mx) : 0.f;
  float pn = __expf(simn - mx);
  float s  = p;
#pragma unroll
  for (int off = 16; off >= 1; off >>= 1)
    s += __shfl_xor(s, off, 32);
  s += pn;
  const float inv = 1.f / s;

  sp[h][lane] = p * inv;
  si[h][lane] = idx;
  __syncthreads();

  // lane now owns output dims d0 = 2*lane, d0+1
  const int d0 = 2 * lane;
  const float pw = pn * inv;
  float acc0 = pw * nv[d0];
  float acc1 = pw * nv[d0 + 1];
#pragma unroll 4
  for (int a = 0; a < AA; ++a) {
    const float pa = sp[h][a];
    const float* __restrict__ vrow = KV + (size_t)si[h][a] * KVW + (HH * DHH) + h * DHH;
    acc0 += pa * vrow[d0];
    acc1 += pa * vrow[d0 + 1];
  }

  float* __restrict__ orow = out + (size_t)n * DD + h * DHH;
  orow[d0]     = acc0;
  orow[d0 + 1] = acc1;
}

// ---------------------------------------------------------------------------
extern "C" void kernel_launch(void* const* d_in, const int* in_sizes, int n_in,
                              void* d_out, int out_size, void* d_ws, size_t ws_size,
                              hipStream_t stream) {
  const float*         x      = (const float*)d_in[0];
  const int*           adj    = (const int*)d_in[1];
  const unsigned char* mask   = (const unsigned char*)d_in[2];
  const float*         Wq     = (const float*)d_in[3];
  const float*         Wkv    = (const float*)d_in[4];
  const float*         Wo     = (const float*)d_in[5];
  const float*         bo     = (const float*)d_in[6];
  const float*         null_k = (const float*)d_in[7];
  const float*         null_v = (const float*)d_in[8];
  float*               out    = (float*)d_out;

  const int N = in_sizes[0] / DD;        // 10000

  // Workspace: Q (N*256 f32) | KV (N*512 f32). Attention output aliases Q.
  float* Q  = (float*)d_ws;
  float* KV = Q + (size_t)N * DD;

  const int rtiles = (N + 15) / 16;
  const dim3 blk(128);

  // 1) Q = x @ Wq          (M=256 -> grid.y = 1)
  gemm_wmma_f32<<<dim3(rtiles, 1), blk, 0, stream>>>(x, Wq, nullptr, Q, N, DD, DD);
  // 2) KV = x @ Wkv        (M=512 -> grid.y = 2)  [gather moved AFTER projection]
  gemm_wmma_f32<<<dim3(rtiles, 2), blk, 0, stream>>>(x, Wkv, nullptr, KV, N, DD, KVW);
  // 3) attention: one wave per (n, h); output overwrites Q in place
  adj_attention<<<dim3(N), blk, 0, stream>>>(Q, KV, adj, mask, null_k, null_v, Q, N);
  // 4) out = att @ Wo + bo
  gemm_wmma_f32<<<dim3(rtiles, 1), blk, 0, stream>>>(Q, Wo, bo, out, N, DD, DD);
}